// CDF_quadratic_20031727468853
// MI455X (gfx1250) — compile-verified
//
#include <hip/hip_runtime.h>
#include <math.h>
#include <stdint.h>

#define DIM 8
#define RGEOM 1.2f
#define BOUND_F 10.0f
#define BETA_F 1e-6f
#define LOG_BETA_F -13.815510558f   // logf(1e-6)

typedef __attribute__((ext_vector_type(4))) float v4f;

// Workspace / LDS table layout (dword offsets, each section 16B-aligned)
#define MESH_OFF 0      // 65 floats: mesh[0..64]                    (65..71 = pad)
#define IELM_OFF 72     // 64 floats: 1.0f / elmt[k]
#define PDF_OFF  136    // 8 x 65 floats, [d*65 + k]  (row len 65 ≡ 1 mod 64 -> conflict-free)
#define FREF_OFF 656    // 8 x 64 floats, [d*64 + k]
#define TBL_DWORDS 1168 // total, multiple of 4
#define TBL_VEC4 (TBL_DWORDS / 4)  // 292 x b128 async copies

// ---------------------------------------------------------------------------
// One-block setup kernel: reproduces _mesh_and_elmt() + _pdf_normalize(p)
// into the workspace tables.
// ---------------------------------------------------------------------------
__global__ __launch_bounds__(128) void setup_tables_kernel(
    const float* __restrict__ p, float* __restrict__ ws) {
  __shared__ float mesh_s[65];
  __shared__ float elmt_s[64];
  const int t = threadIdx.x;

  if (t <= 64) {
    float idx = (float)t - 32.0f;
    float x1L = BOUND_F * (RGEOM - 1.0f) / (powf(RGEOM, 32.0f) - 1.0f);
    float a   = fabsf(idx);
    float xr  = (1.0f - powf(RGEOM, a)) / (1.0f - RGEOM);
    xr = (idx >= 0.0f) ? (x1L * xr) : (-x1L * xr);
    xr = (xr + BOUND_F) / (2.0f * BOUND_F);
    mesh_s[t] = (t == 0) ? 0.0f : ((t == 64) ? 1.0f : xr);
  }
  __syncthreads();
  if (t < 64) elmt_s[t] = mesh_s[t + 1] - mesh_s[t];
  __syncthreads();

  if (t <= 64) ws[MESH_OFF + t] = mesh_s[t];
  if (t >= 65 && t < 72) ws[t] = 0.0f;               // pad
  if (t < 64) ws[IELM_OFF + t] = 1.0f / elmt_s[t];   // reciprocal table

  if (t < DIM) {
    const int d = t;
    float ep[63];
    float sum = 0.0f;
    for (int i = 0; i < 63; ++i) {
      float e = expf(p[i * DIM + d]);
      ep[i] = e;
      sum += e * 0.5f * (elmt_s[i] + elmt_s[i + 1]);
    }
    float scale = (1.0f - (elmt_s[0] + elmt_s[63]) * BETA_F * 0.5f) / sum;
    float pdfv[65];
    pdfv[0]  = BETA_F;
    for (int i = 0; i < 63; ++i) pdfv[i + 1] = scale * ep[i];
    pdfv[64] = BETA_F;
    for (int k = 0; k <= 64; ++k) ws[PDF_OFF + d * 65 + k] = pdfv[k];
    float acc = 0.0f;
    for (int k = 0; k < 64; ++k) {                   // F_ref[k] = sum_{i<k} cell[i]
      ws[FREF_OFF + d * 64 + k] = acc;
      acc += 0.5f * (pdfv[k] + pdfv[k + 1]) * elmt_s[k];
    }
  }
}

// ---------------------------------------------------------------------------
// Main kernel: one thread per point (8 dims). Tables staged to LDS via the
// CDNA5 async global->LDS copy path (ASYNCcnt), then pure streaming NT B128
// I/O. ~144 MB of traffic total -> HBM-bound (~6.2 us at 23.3 TB/s).
// ---------------------------------------------------------------------------
__global__ __launch_bounds__(256) void cdf_quadratic_kernel(
    const float* __restrict__ x, const float* __restrict__ logdet_in,
    const float* __restrict__ tbl, float* __restrict__ y_out,
    float* __restrict__ logdet_out, int n) {
  __shared__ float smem[TBL_DWORDS];

  // Async-stage the 4.6KB table block into LDS: each active lane copies 16B
  // directly into LDS (no VGPR round trip). Low 32 bits of the generic
  // shared pointer are the LDS byte offset.
  for (int i = threadIdx.x; i < TBL_VEC4; i += 256) {
    unsigned lds_off = (unsigned)(uintptr_t)(&smem[i * 4]);
    unsigned long long gaddr = (unsigned long long)(uintptr_t)(tbl + i * 4);
    asm volatile("global_load_async_to_lds_b128 %0, %1, off"
                 :: "v"(lds_off), "v"(gaddr) : "memory");
  }
  asm volatile("s_wait_asynccnt 0x0" ::: "memory");
  __syncthreads();

  int i = blockIdx.x * 256 + threadIdx.x;
  if (i >= n) return;

  const v4f* xv = (const v4f*)x + (size_t)i * 2;
  v4f xa = __builtin_nontemporal_load(&xv[0]);
  v4f xb = __builtin_nontemporal_load(&xv[1]);
  float xr[8] = {xa.x, xa.y, xa.z, xa.w, xb.x, xb.y, xb.z, xb.w};
  float yv[8];
  float s = 0.0f;
  float dprod0 = 1.0f, dprod1 = 1.0f;

#pragma unroll
  for (int d = 0; d < 8; ++d) {
    float u = (xr[d] + BOUND_F) * (1.0f / (2.0f * BOUND_F));

    // k = searchsorted(mesh, u, 'right') - 1 over k in [0,63].
    // mesh[0]=0 and mesh[64]=1 exactly, so cover = (u >= 0) && (u < 1) and the
    // search only needs 6 probes over indices 0..63.
    int pos = 0;
#pragma unroll
    for (int step = 32; step >= 1; step >>= 1) {
      int np = pos + step;
      if (np <= 63 && smem[MESH_OFF + np] <= u) pos = np;
    }
    bool cover = (u >= 0.0f) && (u < 1.0f);
    int  kc    = pos;

    float v1  = smem[PDF_OFF  + d * 65 + kc];
    float v2  = smem[PDF_OFF  + d * 65 + kc + 1];
    float xm  = u - smem[MESH_OFF + kc];
    float ih  = smem[IELM_OFF + kc];
    float Fp  = smem[FREF_OFF + d * 64 + kc];

    float islope = (v2 - v1) * ih;
    float yy  = cover ? (Fp + xm * (0.5f * xm * islope + v1)) : u;
    float dld = cover ? (xm * islope + v1) : 1.0f;
    if (d < 4) dprod0 *= dld; else dprod1 *= dld;   // 4-way products stay normal (>=1e-24)

    yy = yy * (2.0f * BOUND_F) - BOUND_F;
    if (yy > BOUND_F)  { yy = BETA_F * (yy - BOUND_F) + BOUND_F; s += LOG_BETA_F; }
    if (yy < -BOUND_F) { yy = BETA_F * (yy + BOUND_F) - BOUND_F; s += LOG_BETA_F; }
    yv[d] = yy;
  }
  s += __logf(dprod0) + __logf(dprod1);

  v4f* yo = (v4f*)y_out + (size_t)i * 2;
  v4f o0 = {yv[0], yv[1], yv[2], yv[3]};
  v4f o1 = {yv[4], yv[5], yv[6], yv[7]};
  __builtin_nontemporal_store(o0, &yo[0]);
  __builtin_nontemporal_store(o1, &yo[1]);
  float ldin = __builtin_nontemporal_load(logdet_in + i);
  __builtin_nontemporal_store(ldin + s, logdet_out + i);
}

// ---------------------------------------------------------------------------
extern "C" void kernel_launch(void* const* d_in, const int* in_sizes, int n_in,
                              void* d_out, int out_size, void* d_ws, size_t ws_size,
                              hipStream_t stream) {
  const float* x      = (const float*)d_in[0];   // [N, 8]
  const float* logdet = (const float*)d_in[1];   // [N, 1]
  const float* p      = (const float*)d_in[2];   // [63, 8]
  const int n = in_sizes[0] / DIM;

  float* ws  = (float*)d_ws;
  float* y   = (float*)d_out;                    // [N, 8]
  float* ld  = (float*)d_out + (size_t)n * DIM;  // [N, 1]

  setup_tables_kernel<<<1, 128, 0, stream>>>(p, ws);
  const int blocks = (n + 255) / 256;
  cdf_quadratic_kernel<<<blocks, 256, 0, stream>>>(x, logdet, ws, y, ld, n);
}